// Cross_Mamba_Attention_SSM_51780125720655
// MI455X (gfx1250) — compile-verified
//
#include <hip/hip_runtime.h>
#include <math.h>

// ---------------- problem constants ----------------
#define L_SEQ   8192
#define BATCH   2
#define DIN     384        // D_INNER
#define NST     16         // D_STATE
#define RNK     12         // DT_RANK
#define NPROJ   416        // DIN (dt) + NST (B) + NST (C)
#define M_TOT   (BATCH * L_SEQ)   // 16384 flattened (b,l) rows
#define NCHUNK  64
#define CLEN    (L_SEQ / NCHUNK)  // 128
#define NTILE_N (NPROJ / 16)      // 26
#define APAD    (DIN + 4)         // LDS row stride: 388 % 64 == 4 -> conflict-free

typedef __attribute__((ext_vector_type(2))) float v2f;
typedef __attribute__((ext_vector_type(8))) float v8f;

__device__ __forceinline__ float softplus_f(float x) {
  return (x > 20.0f) ? x : log1pf(__expf(x));
}

// LDS byte offset of a __shared__ object (wave-relative address for async ops)
__device__ __forceinline__ unsigned lds_offset(const void* p) {
  return (unsigned)(uintptr_t)(__attribute__((address_space(3))) const void*)p;
}

// async global -> LDS copy of 16 bytes per lane (ASYNCcnt tracked)
__device__ __forceinline__ void async_load_b128(unsigned lds_off, const void* gaddr) {
  asm volatile("global_load_async_to_lds_b128 %0, %1, off"
               :: "v"(lds_off), "v"(gaddr) : "memory");
}

__device__ __forceinline__ void wait_async0() {
  asm volatile("s_wait_asynccnt 0" ::: "memory");
}

// ---------------------------------------------------------------------------
// Kernel 1: build combined weights per stream:
//   rows [0,384)   : W_dt[d][k] = sum_r dt_proj_w[d][r] * x_proj_w[r][k]
//   rows [384,400) : x_proj_w[12+n][k]   (B rows)
//   rows [400,416) : x_proj_w[28+n][k]   (C rows)
// ---------------------------------------------------------------------------
__global__ __launch_bounds__(256) void build_wcomb_kernel(
    const float* __restrict__ xw1, const float* __restrict__ xw2,
    const float* __restrict__ dtw1, const float* __restrict__ dtw2,
    float* __restrict__ wcomb) {
  int t = blockIdx.x * blockDim.x + threadIdx.x;
  const int total = 2 * NPROJ * DIN;
  if (t >= total) return;
  int s = t / (NPROJ * DIN);
  int rem = t - s * (NPROJ * DIN);
  int row = rem / DIN;
  int k = rem - row * DIN;
  const float* xw  = s ? xw2 : xw1;
  const float* dtw = s ? dtw2 : dtw1;
  float v;
  if (row < DIN) {
    v = 0.0f;
#pragma unroll
    for (int r = 0; r < RNK; ++r)
      v = fmaf(dtw[row * RNK + r], xw[r * DIN + k], v);
  } else {
    v = xw[(RNK + (row - DIN)) * DIN + k];
  }
  wcomb[t] = v;
}

// ---------------------------------------------------------------------------
// Kernel 2: WMMA f32 GEMM.  Y[m, n] = sum_k X[m,k] * Wcomb[n,k]
// Block = 256 threads (8 waves) owns one 16-row M tile; the 16x384 X tile
// (contiguous 24KB) is staged once into LDS via global_load_async_to_lds_b128,
// then each wave computes 3-4 of the 26 N tiles with
// V_WMMA_F32_16X16X4_F32, reusing the LDS A fragment across its tiles.
// A frag (16x4 f32): lane j(g) holds A[m0+j][k+2g .. k+2g+1] (from LDS).
// B frag (4x16 f32): lane j(g) holds W[n0+j][k+2g .. k+2g+1] (global, L0-hot).
// C/D: VGPR r, lane (g,j) = D[r + 8g][j].
// dt columns stored transposed (s,d,m) for the scan; B/C stored (m,16).
// ---------------------------------------------------------------------------
__device__ __forceinline__ void store_tile(
    const v8f& acc, int tileN, int s, int m0, int g, int j,
    float* __restrict__ dtws, float* __restrict__ bbuf, float* __restrict__ cbuf) {
  if (tileN < 24) {                       // dt region -> transposed (s,d,m)
    int n = tileN * 16 + j;
    float* dst = dtws + ((size_t)(s * DIN + n)) * M_TOT + m0 + 8 * g;
    float4 lo = make_float4(acc[0], acc[1], acc[2], acc[3]);
    float4 hi = make_float4(acc[4], acc[5], acc[6], acc[7]);
    *(float4*)(dst)     = lo;
    *(float4*)(dst + 4) = hi;
  } else {                                // B (tile 24) / C (tile 25) regions
    float* base = (tileN == 24) ? bbuf : cbuf;
    float* dst = base + (size_t)s * M_TOT * NST + ((size_t)(m0 + 8 * g)) * NST + j;
#pragma unroll
    for (int r = 0; r < 8; ++r) dst[(size_t)r * NST] = acc[r];
  }
}

template <int NT>
__device__ __forceinline__ void gemm_tiles(
    const float* ldsA, const float* __restrict__ Wt,
    int w, int j, int g, int s, int m0,
    float* __restrict__ dtws, float* __restrict__ bbuf, float* __restrict__ cbuf) {
  const float* brow[NT];
#pragma unroll
  for (int i = 0; i < NT; ++i) {
    int tn = w + 8 * i;
    brow[i] = Wt + (size_t)(tn * 16 + j) * DIN + 2 * g;
  }
  v8f acc[NT];
#pragma unroll
  for (int i = 0; i < NT; ++i) acc[i] = {};
  for (int k = 0; k < DIN; k += 4) {
    v2f a = *(const v2f*)(ldsA + k);          // ds_load_b64, conflict-free
#pragma unroll
    for (int i = 0; i < NT; ++i) {
      v2f b = *(const v2f*)(brow[i] + k);
      acc[i] = __builtin_amdgcn_wmma_f32_16x16x4_f32(false, a, false, b,
                                                     (short)0, acc[i], false, false);
    }
  }
#pragma unroll
  for (int i = 0; i < NT; ++i)
    store_tile(acc[i], w + 8 * i, s, m0, g, j, dtws, bbuf, cbuf);
}

__global__ __launch_bounds__(256) void proj_gemm_kernel(
    const float* __restrict__ x1, const float* __restrict__ x2,
    const float* __restrict__ wcomb,
    float* __restrict__ dtws, float* __restrict__ bbuf, float* __restrict__ cbuf) {
  __shared__ float tileA[16 * APAD];          // padded 16x384 tile (~24.8KB)
  const int tid = threadIdx.x;
  const int lane = tid & 31;
  const int w = tid >> 5;                     // wave id 0..7
  const int j = lane & 15;
  const int g = lane >> 4;
  const int m0 = blockIdx.x * 16;
  const int s = blockIdx.y;

  const float* X  = s ? x2 : x1;
  const float* Wt = wcomb + (size_t)s * NPROJ * DIN;

  // ---- stage the contiguous 24KB X tile into LDS (async, 16B/lane) ----
  {
    const char* gsrc = (const char*)(X + (size_t)m0 * DIN);
    const unsigned lbase = lds_offset(tileA);
#pragma unroll
    for (int i = 0; i < 6; ++i) {
      int chunk = tid + 256 * i;              // 1536 x 16B chunks
      int row = chunk / 96;                   // 96 chunks per 1536B row
      int col = chunk - row * 96;
      async_load_b128(lbase + (unsigned)(row * APAD * 4 + col * 16),
                      gsrc + (size_t)row * (DIN * 4) + (size_t)col * 16);
    }
    wait_async0();
  }
  __syncthreads();

  const float* ldsA = &tileA[j * APAD + 2 * g];
  if (w < 2)
    gemm_tiles<4>(ldsA, Wt, w, j, g, s, m0, dtws, bbuf, cbuf);   // tiles w,w+8,w+16,w+24
  else
    gemm_tiles<3>(ldsA, Wt, w, j, g, s, m0, dtws, bbuf, cbuf);   // tiles w,w+8,w+16
}

// ---------------------------------------------------------------------------
// Kernel 3: scan phase 1 — per-chunk recurrence with h0 = 0.
// wave = 2 channels x 16 states; lane (c=lane>>4, n=lane&15).
// Stores chunk-end state E and decay product P = exp(A * sum(softplus dt)).
// ---------------------------------------------------------------------------
__global__ __launch_bounds__(32) void scan_chunk_kernel(
    const float* __restrict__ x1, const float* __restrict__ x2,
    const float* __restrict__ alog1, const float* __restrict__ alog2,
    const float* __restrict__ dtb1, const float* __restrict__ dtb2,
    const float* __restrict__ dtws, const float* __restrict__ bbuf,
    float* __restrict__ hend, float* __restrict__ aprod) {
  const int lane = threadIdx.x;
  const int j = lane & 15;
  const int c = lane >> 4;
  const int d = blockIdx.x * 2 + c;
  const int chunk = blockIdx.y;
  const int s = blockIdx.z >> 1;
  const int b = blockIdx.z & 1;

  const float* xs = s ? x2 : x1;
  const float Av   = -__expf((s ? alog2 : alog1)[d * NST + j]);
  const float bias = (s ? dtb2 : dtb1)[d];

  const int l0 = chunk * CLEN;
  const float* dtp = dtws + ((size_t)(s * DIN + d)) * M_TOT + b * L_SEQ + l0;
  const float* up  = xs + ((size_t)(b * L_SEQ + l0)) * DIN + d;
  const float* Bp  = bbuf + ((size_t)s * M_TOT + (size_t)b * L_SEQ + l0) * NST + j;

  float h = 0.0f, dtsum = 0.0f;
  for (int t = 0; t < CLEN; ++t) {
    if ((t & 7) == 0) __builtin_prefetch(up + (size_t)(t + 8) * DIN, 0, 0);
    float dtv = softplus_f(dtp[t] + bias);
    float uv = up[(size_t)t * DIN];
    float Bv = Bp[(size_t)t * NST];
    float dA = __expf(dtv * Av);
    h = fmaf(dA, h, dtv * uv * Bv);
    dtsum += dtv;
  }
  size_t idx = ((((size_t)s * BATCH + b) * DIN + d) * NCHUNK + chunk) * NST + j;
  hend[idx]  = h;
  aprod[idx] = __expf(dtsum * Av);
}

// ---------------------------------------------------------------------------
// Kernel 4: scan phase 2 — serial fix-up across the 64 chunks per chain.
//   h_init[0] = 0 ; h_init[c] = P[c-1]*h_init[c-1] + E[c-1]
// ---------------------------------------------------------------------------
__global__ __launch_bounds__(256) void chunk_fixup_kernel(
    const float* __restrict__ hend, const float* __restrict__ aprod,
    float* __restrict__ hinit) {
  int t = blockIdx.x * blockDim.x + threadIdx.x;
  const int total = 2 * BATCH * DIN * NST;
  if (t >= total) return;
  int j = t & 15;
  int rest = t >> 4;
  int d = rest % DIN;
  int sb = rest / DIN;                        // s*BATCH + b
  size_t base = (((size_t)sb * DIN + d) * NCHUNK) * NST + j;
  float h = 0.0f;
  for (int c2 = 0; c2 < NCHUNK; ++c2) {
    size_t id = base + (size_t)c2 * NST;
    float he = hend[id];
    float ap = aprod[id];
    hinit[id] = h;
    h = fmaf(ap, h, he);
  }
}

// ---------------------------------------------------------------------------
// Kernel 5: scan phase 3 — re-run each chunk from correct h_init, emit
//   y[m,d] = sum_n h*C_other + u*D   (C is taken from the OTHER stream).
// ---------------------------------------------------------------------------
__global__ __launch_bounds__(32) void scan_final_kernel(
    const float* __restrict__ x1, const float* __restrict__ x2,
    const float* __restrict__ alog1, const float* __restrict__ alog2,
    const float* __restrict__ dtb1, const float* __restrict__ dtb2,
    const float* __restrict__ Dv1, const float* __restrict__ Dv2,
    const float* __restrict__ dtws, const float* __restrict__ bbuf,
    const float* __restrict__ cbuf, const float* __restrict__ hinit,
    float* __restrict__ ybuf) {
  const int lane = threadIdx.x;
  const int j = lane & 15;
  const int c = lane >> 4;
  const int d = blockIdx.x * 2 + c;
  const int chunk = blockIdx.y;
  const int s = blockIdx.z >> 1;
  const int b = blockIdx.z & 1;

  const float* xs = s ? x2 : x1;
  const float Av   = -__expf((s ? alog2 : alog1)[d * NST + j]);
  const float bias = (s ? dtb2 : dtb1)[d];
  const float Dd   = (s ? Dv2 : Dv1)[d];

  const int l0 = chunk * CLEN;
  const float* dtp = dtws + ((size_t)(s * DIN + d)) * M_TOT + b * L_SEQ + l0;
  const float* up  = xs + ((size_t)(b * L_SEQ + l0)) * DIN + d;
  const float* Bp  = bbuf + ((size_t)s * M_TOT + (size_t)b * L_SEQ + l0) * NST + j;
  const float* Cp  = cbuf + ((size_t)(s ^ 1) * M_TOT + (size_t)b * L_SEQ + l0) * NST + j;
  float* yp = ybuf + ((size_t)s * M_TOT + (size_t)b * L_SEQ + l0) * DIN + d;

  size_t idx = ((((size_t)s * BATCH + b) * DIN + d) * NCHUNK + chunk) * NST + j;
  float h = hinit[idx];

  for (int t = 0; t < CLEN; ++t) {
    if ((t & 7) == 0) __builtin_prefetch(up + (size_t)(t + 8) * DIN, 0, 0);
    float dtv = softplus_f(dtp[t] + bias);
    float uv = up[(size_t)t * DIN];
    float Bv = Bp[(size_t)t * NST];
    float Cv = Cp[(size_t)t * NST];
    float dA = __expf(dtv * Av);
    h = fmaf(dA, h, dtv * uv * Bv);
    float yv = h * Cv;
    // reduce over the 16 states (half-wave butterfly)
    yv += __shfl_xor(yv, 1, 16);
    yv += __shfl_xor(yv, 2, 16);
    yv += __shfl_xor(yv, 4, 16);
    yv += __shfl_xor(yv, 8, 16);
    if (j == 0) yp[(size_t)t * DIN] = fmaf(uv, Dd, yv);
  }
}

// ---------------------------------------------------------------------------
// Kernel 6: LayerNorm over d (=384) per (s,b,l) row; write final output.
// 192 threads (6 waves), 2 elements/thread.
// ---------------------------------------------------------------------------
__global__ __launch_bounds__(192) void layernorm_kernel(
    const float* __restrict__ ybuf,
    const float* __restrict__ ln1g, const float* __restrict__ ln1b,
    const float* __restrict__ ln2g, const float* __restrict__ ln2b,
    float* __restrict__ out) {
  __shared__ float ssum[6];
  __shared__ float ssq[6];
  __shared__ float sstat[2];
  const int tid = threadIdx.x;
  const int l = blockIdx.x, b = blockIdx.y, s = blockIdx.z;
  const float* row = ybuf + ((size_t)s * M_TOT + (size_t)b * L_SEQ + l) * DIN;
  float v0 = row[tid];
  float v1 = row[tid + 192];
  float sum = v0 + v1;
  float sq = fmaf(v0, v0, v1 * v1);
#pragma unroll
  for (int m = 16; m >= 1; m >>= 1) {
    sum += __shfl_xor(sum, m, 32);
    sq  += __shfl_xor(sq, m, 32);
  }
  int wid = tid >> 5;
  if ((tid & 31) == 0) { ssum[wid] = sum; ssq[wid] = sq; }
  __syncthreads();
  if (tid == 0) {
    float S = 0.0f, Q = 0.0f;
#pragma unroll
    for (int i = 0; i < 6; ++i) { S += ssum[i]; Q += ssq[i]; }
    float mu = S * (1.0f / DIN);
    float var = Q * (1.0f / DIN) - mu * mu;
    sstat[0] = mu;
    sstat[1] = rsqrtf(var + 1e-5f);
  }
  __syncthreads();
  float mu = sstat[0], rs = sstat[1];
  const float* gg = s ? ln2g : ln1g;
  const float* bb = s ? ln2b : ln1b;
  float* o = out + (size_t)s * M_TOT * DIN + ((size_t)b * L_SEQ + l) * DIN;
  o[tid]       = fmaf((v0 - mu) * rs, gg[tid],       bb[tid]);
  o[tid + 192] = fmaf((v1 - mu) * rs, gg[tid + 192], bb[tid + 192]);
}

// ---------------------------------------------------------------------------
extern "C" void kernel_launch(void* const* d_in, const int* in_sizes, int n_in,
                              void* d_out, int out_size, void* d_ws, size_t ws_size,
                              hipStream_t stream) {
  (void)in_sizes; (void)n_in; (void)out_size; (void)ws_size;
  const float* x1    = (const float*)d_in[0];
  const float* x2    = (const float*)d_in[1];
  const float* xw1   = (const float*)d_in[2];
  const float* xw2   = (const float*)d_in[3];
  const float* dtw1  = (const float*)d_in[4];
  const float* dtb1  = (const float*)d_in[5];
  const float* dtw2  = (const float*)d_in[6];
  const float* dtb2  = (const float*)d_in[7];
  const float* alog1 = (const float*)d_in[8];
  const float* alog2 = (const float*)d_in[9];
  const float* Dv1   = (const float*)d_in[10];
  const float* Dv2   = (const float*)d_in[11];
  const float* ln1g  = (const float*)d_in[12];
  const float* ln1b  = (const float*)d_in[13];
  const float* ln2g  = (const float*)d_in[14];
  const float* ln2b  = (const float*)d_in[15];
  float* out = (float*)d_out;

  // workspace layout (floats) — ~125 MB total, fits L2 with margin
  float* wsf   = (float*)d_ws;
  float* wcomb = wsf;                                                // 2*416*384
  float* dtws  = wcomb + (size_t)2 * NPROJ * DIN;                    // 2*384*16384
  float* bbuf  = dtws  + (size_t)2 * DIN * M_TOT;                    // 2*16384*16
  float* cbuf  = bbuf  + (size_t)2 * M_TOT * NST;                    // 2*16384*16
  float* hend  = cbuf  + (size_t)2 * M_TOT * NST;                    // 2*2*384*64*16
  float* aprod = hend  + (size_t)2 * BATCH * DIN * NCHUNK * NST;
  float* hinit = aprod + (size_t)2 * BATCH * DIN * NCHUNK * NST;
  float* ybuf  = hinit + (size_t)2 * BATCH * DIN * NCHUNK * NST;     // 2*16384*384

  build_wcomb_kernel<<<dim3((2 * NPROJ * DIN + 255) / 256), 256, 0, stream>>>(
      xw1, xw2, dtw1, dtw2, wcomb);

  proj_gemm_kernel<<<dim3(M_TOT / 16, 2, 1), 256, 0, stream>>>(
      x1, x2, wcomb, dtws, bbuf, cbuf);

  scan_chunk_kernel<<<dim3(DIN / 2, NCHUNK, 4), 32, 0, stream>>>(
      x1, x2, alog1, alog2, dtb1, dtb2, dtws, bbuf, hend, aprod);

  chunk_fixup_kernel<<<dim3((2 * BATCH * DIN * NST + 255) / 256), 256, 0, stream>>>(
      hend, aprod, hinit);

  scan_final_kernel<<<dim3(DIN / 2, NCHUNK, 4), 32, 0, stream>>>(
      x1, x2, alog1, alog2, dtb1, dtb2, Dv1, Dv2, dtws, bbuf, cbuf, hinit, ybuf);

  layernorm_kernel<<<dim3(L_SEQ, BATCH, 2), 192, 0, stream>>>(
      ybuf, ln1g, ln1b, ln2g, ln2b, out);
}